// InjectedSpectrum_44126493999489
// MI455X (gfx1250) — compile-verified
//
#include <hip/hip_runtime.h>
#include <math.h>

// ---------------------------------------------------------------------------
// Blocked right-looking backward cascade solve for gfx1250 (MI455X).
//   Phase A: 1 wave solves the 64-row diagonal block sequentially from LDS.
//   Phase B: all 32 workgroups apply the block's rank-192 update to all lower
//            rows via V_WMMA_F32_16X16X4_F32 (f32 kept: stiff recurrence,
//            and we're latency-bound, not HBM-bound: K = 151 MB -> 6.5us floor).
// ---------------------------------------------------------------------------

#define NE 2048
#define NX 3
#define BS 64
#define NB 32               // NE-1 = 2047 rows: block 0 has 63, rest 64
#define NWG 32
#define NTHREADS 256
#define NWAVES (NWG * (NTHREADS / 32))

typedef float v2f __attribute__((ext_vector_type(2)));
typedef float v8f __attribute__((ext_vector_type(8)));

__global__ void bar_init_kernel(unsigned* bar) {
  if (threadIdx.x < 2) bar[threadIdx.x] = 0u;
}

__device__ __forceinline__ void grid_sync(unsigned* cnt, unsigned* gen) {
  __threadfence();
  __syncthreads();
  if (threadIdx.x == 0) {
    unsigned g = __hip_atomic_load(gen, __ATOMIC_RELAXED, __HIP_MEMORY_SCOPE_AGENT);
    unsigned a = __hip_atomic_fetch_add(cnt, 1u, __ATOMIC_ACQ_REL, __HIP_MEMORY_SCOPE_AGENT);
    if (a == NWG - 1) {
      __hip_atomic_store(cnt, 0u, __ATOMIC_RELAXED, __HIP_MEMORY_SCOPE_AGENT);
      __hip_atomic_fetch_add(gen, 1u, __ATOMIC_ACQ_REL, __HIP_MEMORY_SCOPE_AGENT);
    } else {
      while (__hip_atomic_load(gen, __ATOMIC_ACQUIRE, __HIP_MEMORY_SCOPE_AGENT) == g)
        __builtin_amdgcn_s_sleep(8);
    }
  }
  __syncthreads();
  __threadfence();
}

__global__ void __launch_bounds__(NTHREADS, 1)
cascade_kernel(const float* __restrict__ Eg, const float* __restrict__ R,
               const float* __restrict__ K, const float* __restrict__ S0,
               const float* __restrict__ SC, float* __restrict__ out,
               float* __restrict__ ws)
{
  __shared__ __align__(16) float tileK[9 * BS * BS];   // 147456 B diagonal tile
  __shared__ float vblk[3 * BS];                       // v[p, j] for current block

  float* accG  = ws;                 // [3][NE] running sums from finished blocks
  float* vG    = ws + 3 * NE;        // [3][NE] v[p,j] = w[j]*F[p,j] (+src_last col)
  float* FG    = ws + 6 * NE;        // [3][NE] raw F
  unsigned* bar = (unsigned*)(ws + 9 * NE);

  const int tid  = threadIdx.x;
  const int lane = tid & 31;
  const int gt   = blockIdx.x * NTHREADS + tid;
  const int gw   = gt >> 5;

  const float dy = logf(Eg[NE - 1] / Eg[0]) / (float)(NE - 1);

  // ---- init: zero acc; seed column j = NE-1 (F, v incl. src_last fold) ----
  for (int t = gt; t < 3 * NE; t += NWG * NTHREADS) accG[t] = 0.0f;
  if (blockIdx.x == 0 && tid == 0) {
    float sl[3], fx[3];
    for (int p = 0; p < 3; ++p) sl[p] = S0[p] / R[p * NE + (NE - 1)];
    for (int x = 0; x < 3; ++x) {
      float kk = 0.0f;
      for (int p = 0; p < 3; ++p)
        kk += K[(((size_t)(x * 3 + p)) * NE + (NE - 1)) * NE + (NE - 1)] * sl[p];
      fx[x] = (SC[x * NE + (NE - 1)] + kk) / R[x * NE + (NE - 1)];
    }
    for (int x = 0; x < 3; ++x) {
      FG[x * NE + (NE - 1)] = fx[x];
      vG[x * NE + (NE - 1)] = 0.5f * dy * Eg[NE - 1] * fx[x] + sl[x];
    }
  }
  grid_sync(bar, bar + 1);

  for (int b = 0; b < NB; ++b) {
    const int lo = (NB - 1 - b) * BS;   // block rows/cols [lo, lo+63]

    // ================= Phase A: sequential diagonal solve (WG0) ============
    if (blockIdx.x == 0) {
      for (int t = tid; t < 9 * BS * (BS / 4); t += NTHREADS) {
        const int xp  = t / (BS * (BS / 4));
        const int rem = t - xp * (BS * (BS / 4));
        const int r   = rem >> 4;
        const int q   = rem & 15;
        const float4* srcp =
            reinterpret_cast<const float4*>(K + ((size_t)xp * NE + (lo + r)) * NE + lo);
        reinterpret_cast<float4*>(tileK)[t] = srcp[q];
      }
      if (b == 0 && tid < 3)  // block 0's j-range includes the fixed col 2047
        vblk[tid * BS + (BS - 1)] = vG[(size_t)tid * NE + (NE - 1)];
      __syncthreads();

      if (tid < 32) {
        volatile float* vb = vblk;
        const int ihi = lo + ((b == 0) ? (BS - 2) : (BS - 1));
        const int jhi = lo + BS - 1;
        for (int i = ihi; i >= lo; --i) {
          const int ri = i - lo;
          float p0 = 0.f, p1 = 0.f, p2 = 0.f;
          for (int j = i + 1 + lane; j <= jhi; j += 32) {
            const int c = j - lo;
            const float w0 = vb[0 * BS + c];
            const float w1 = vb[1 * BS + c];
            const float w2 = vb[2 * BS + c];
            p0 += tileK[(0*BS+ri)*BS+c]*w0 + tileK[(1*BS+ri)*BS+c]*w1 + tileK[(2*BS+ri)*BS+c]*w2;
            p1 += tileK[(3*BS+ri)*BS+c]*w0 + tileK[(4*BS+ri)*BS+c]*w1 + tileK[(5*BS+ri)*BS+c]*w2;
            p2 += tileK[(6*BS+ri)*BS+c]*w0 + tileK[(7*BS+ri)*BS+c]*w1 + tileK[(8*BS+ri)*BS+c]*w2;
          }
          for (int off = 16; off > 0; off >>= 1) {
            p0 += __shfl_down(p0, off, 32);
            p1 += __shfl_down(p1, off, 32);
            p2 += __shfl_down(p2, off, 32);
          }
          if (lane == 0) {
            const float a0 = SC[0*NE+i] + accG[0*NE+i] + p0;
            const float a1 = SC[1*NE+i] + accG[1*NE+i] + p1;
            const float a2 = SC[2*NE+i] + accG[2*NE+i] + p2;
            const float s  = 0.5f * dy * Eg[i];
            const float b00 = R[0*NE+i] - s * tileK[(0*BS+ri)*BS+ri];
            const float b01 =           - s * tileK[(1*BS+ri)*BS+ri];
            const float b02 =           - s * tileK[(2*BS+ri)*BS+ri];
            const float b10 =           - s * tileK[(3*BS+ri)*BS+ri];
            const float b11 = R[1*NE+i] - s * tileK[(4*BS+ri)*BS+ri];
            const float b12 =           - s * tileK[(5*BS+ri)*BS+ri];
            const float b20 =           - s * tileK[(6*BS+ri)*BS+ri];
            const float b21 =           - s * tileK[(7*BS+ri)*BS+ri];
            const float b22 = R[2*NE+i] - s * tileK[(8*BS+ri)*BS+ri];
            const float m00 = b11*b22 - b12*b21;
            const float m01 = b10*b22 - b12*b20;
            const float m02 = b10*b21 - b11*b20;
            const float inv = 1.0f / (b00*m00 - b01*m01 + b02*m02);
            const float f0 = (a0*m00 - b01*(a1*b22 - b12*a2) + b02*(a1*b21 - b11*a2)) * inv;
            const float f1 = (b00*(a1*b22 - b12*a2) - a0*m01 + b02*(b10*a2 - a1*b20)) * inv;
            const float f2 = (b00*(b11*a2 - a1*b21) - b01*(b10*a2 - a1*b20) + a0*m02) * inv;
            FG[0*NE+i] = f0; FG[1*NE+i] = f1; FG[2*NE+i] = f2;
            const float wv = dy * Eg[i];
            vG[0*NE+i] = wv*f0; vG[1*NE+i] = wv*f1; vG[2*NE+i] = wv*f2;
            vb[0*BS+ri] = wv*f0; vb[1*BS+ri] = wv*f1; vb[2*BS+ri] = wv*f2;
          }
          __builtin_amdgcn_wave_barrier();
          asm volatile("s_wait_dscnt 0" ::: "memory");
          __builtin_amdgcn_wave_barrier();
        }
      }
      __syncthreads();
    }
    grid_sync(bar, bar + 1);

    // ====== Phase B: acc[x,i'<lo] += K[x,p,i',blk] . v[p,blk] via WMMA =====
    const int nt = lo >> 4;            // 16-row tiles below the block
    if (nt > 0) {
      for (int t = tid; t < 3 * BS; t += NTHREADS)
        vblk[t] = vG[(size_t)(t >> 6) * NE + lo + (t & 63)];
      __syncthreads();
      const int koff = (lane >= 16) ? 2 : 0;
      const int mrow = lane & 15;
      for (int pr = gw; pr < nt * 3; pr += NWAVES) {
        const int tile = pr % nt;
        const int x    = pr / nt;
        const int row  = tile * 16 + mrow;
        v8f c = {0.f, 0.f, 0.f, 0.f, 0.f, 0.f, 0.f, 0.f};
        for (int p = 0; p < 3; ++p) {
          const float* krow = K + (((size_t)(x * 3 + p)) * NE + row) * NE + lo + koff;
          __builtin_prefetch(krow, 0, 3);
          const float* vrow = vblk + p * BS + koff;
#pragma unroll
          for (int jc = 0; jc < 16; ++jc) {
            v2f A; A.x = krow[4 * jc]; A.y = krow[4 * jc + 1];
            const float l0 = vrow[4 * jc];
            const float l1 = vrow[4 * jc + 1];
            const bool head = (mrow == 0);       // lanes 0 and 16 feed B col 0
            v2f Bm; Bm.x = head ? l0 : 0.0f; Bm.y = head ? l1 : 0.0f;
            c = __builtin_amdgcn_wmma_f32_16x16x4_f32(false, A, false, Bm,
                                                      (short)0, c, false, false);
          }
        }
        if (mrow == 0) {                          // col 0: lane0 rows 0-7, lane16 rows 8-15
          const int rb = tile * 16 + ((lane == 16) ? 8 : 0);
#pragma unroll
          for (int r2 = 0; r2 < 8; ++r2)
            accG[(size_t)x * NE + rb + r2] += c[r2];
        }
      }
    }
    grid_sync(bar, bar + 1);
  }

  // ---- emit output: row 0 = E_grid, rows 1..3 = clamped F ----
  for (int j = gt; j < NE; j += NWG * NTHREADS) {
    out[j]          = Eg[j];
    out[1 * NE + j] = fmaxf(FG[0 * NE + j], 0.0f);  // float32(1e-200) == 0
    out[2 * NE + j] = fmaxf(FG[1 * NE + j], 0.0f);
    out[3 * NE + j] = fmaxf(FG[2 * NE + j], 0.0f);
  }
}

extern "C" void kernel_launch(void* const* d_in, const int* in_sizes, int n_in,
                              void* d_out, int out_size, void* d_ws, size_t ws_size,
                              hipStream_t stream) {
  (void)in_sizes; (void)n_in; (void)out_size; (void)ws_size;
  const float* Eg = (const float*)d_in[0];
  const float* R  = (const float*)d_in[1];
  const float* K  = (const float*)d_in[2];
  const float* S0 = (const float*)d_in[3];
  const float* SC = (const float*)d_in[4];
  // d_in[5] = T (unused by the reference math)
  float* ws = (float*)d_ws;
  unsigned* bar = (unsigned*)(ws + 9 * NE);
  bar_init_kernel<<<1, 32, 0, stream>>>(bar);
  cascade_kernel<<<NWG, NTHREADS, 0, stream>>>(Eg, R, K, S0, SC, (float*)d_out, ws);
}